// WarpedLoss_18494129177236
// MI455X (gfx1250) — compile-verified
//
#include <hip/hip_runtime.h>

typedef __attribute__((ext_vector_type(2))) float v2f;
typedef __attribute__((ext_vector_type(8))) float v8f;

constexpr int B_ = 8, C_ = 4, H_ = 1024, W_ = 1024, BORDER = 5;
constexpr int IH = H_ - 2 * BORDER;   // 1014
constexpr int IW = W_ - 2 * BORDER;   // 1014
constexpr long long NPIX = (long long)B_ * IH * IW;           // 8,225,568
constexpr float INV_N = 1.0f / ((float)B_ * C_ * IH * IW);    // 1/32,902,272
constexpr int NBLOCKS = 4096;
constexpr int NTHREADS = 256;

__device__ __forceinline__ int reflect_idx(int i, int size) {
    // padding_mode='reflection', align_corners=True over [0, size-1]
    int span = size - 1;              // 1023
    int ai = i < 0 ? -i : i;
    int m = ai % (2 * span);
    return (m > span) ? (2 * span - m) : m;
}

__device__ __forceinline__ void cubic_w(float t, float w[4]) {
    const float A = -0.75f;
    float t1 = t + 1.0f;
    w[0] = ((A * t1 - 5.0f * A) * t1 + 8.0f * A) * t1 - 4.0f * A;
    w[1] = ((A + 2.0f) * t - (A + 3.0f)) * t * t + 1.0f;
    float s = 1.0f - t;
    w[2] = ((A + 2.0f) * s - (A + 3.0f)) * s * s + 1.0f;
    float s2 = 2.0f - t;
    w[3] = ((A * s2 - 5.0f * A) * s2 + 8.0f * A) * s2 - 4.0f * A;
}

// Wave32 sum of `acc` across all lanes using V_WMMA_F32_16X16X4_F32:
// A = lane partials (16x4 tile), B = ones  ->  D rows hold row-sums.
// d[0..7] on lanes 0-15 cover rows 0-7, on lanes 16-31 rows 8-15;
// one xor-16 shuffle combines the two halves. EXEC must be all 1s here.
__device__ __forceinline__ float wave_sum_wmma(float acc) {
    v2f a;  a.x = acc;  a.y = 0.0f;
    v2f bo; bo.x = 1.0f; bo.y = 1.0f;
    v8f c = {};
    v8f d = __builtin_amdgcn_wmma_f32_16x16x4_f32(
        /*neg_a=*/false, a, /*neg_b=*/false, bo,
        /*c_mod=*/(short)0, c, /*reuse_a=*/false, /*reuse_b=*/false);
    float s = 0.0f;
#pragma unroll
    for (int k = 0; k < 8; ++k) s += d[k];
    s += __shfl_xor(s, 16, 32);
    return s;
}

__global__ void __launch_bounds__(NTHREADS)
warp_l1_partial(const float* __restrict__ input, const float* __restrict__ target,
                const float* __restrict__ flow, float* __restrict__ partials) {
    const long long tid    = (long long)blockIdx.x * blockDim.x + threadIdx.x;
    const long long stride = (long long)gridDim.x * blockDim.x;
    const long long plane  = (long long)H_ * W_;

    float acc = 0.0f;
    for (long long p = tid; p < NPIX; p += stride) {
        int xi = (int)(p % IW);
        long long r = p / IW;
        int yi = (int)(r % IH);
        int b  = (int)(r / IH);
        int x = xi + BORDER;
        int y = yi + BORDER;
        int pix = y * W_ + x;

        // flow is streamed exactly once: non-temporal so it can't evict target from L2
        const float* fb = flow + (long long)b * 2 * plane;
        float dx = __builtin_nontemporal_load(fb + pix);
        float dy = __builtin_nontemporal_load(fb + plane + pix);

        float ix = (float)x + dx;
        float iy = (float)y + dy;
        float bxf = floorf(ix), byf = floorf(iy);
        int bx = (int)bxf, by = (int)byf;

        float wx[4], wy[4];
        cubic_w(ix - bxf, wx);
        cubic_w(iy - byf, wy);

        // 16 reflected tap offsets + weights, computed once, reused for 4 channels
        int   off[16];
        float w[16];
#pragma unroll
        for (int j = 0; j < 4; ++j) {
            int Y = reflect_idx(by + (j - 1), H_);
#pragma unroll
            for (int i = 0; i < 4; ++i) {
                off[j * 4 + i] = Y * W_ + reflect_idx(bx + (i - 1), W_);
                w[j * 4 + i]   = wy[j] * wx[i];
            }
        }

        const float* tb = target + (long long)b * C_ * plane;
        const float* ib = input  + (long long)b * C_ * plane;
#pragma unroll
        for (int c = 0; c < C_; ++c) {
            const float* tc = tb + (long long)c * plane;   // L2-resident gather (RT hint)
            float v = 0.0f;
#pragma unroll
            for (int k = 0; k < 16; ++k) v = fmaf(tc[off[k]], w[k], v);
            float in = __builtin_nontemporal_load(ib + (long long)c * plane + pix);
            acc += fabsf(in - v);
        }
    }

    // wave32 reduce via WMMA, then block reduce in LDS (no float atomics -> deterministic)
    float s = wave_sum_wmma(acc);
    __shared__ float wsum[NTHREADS / 32];
    int lane = threadIdx.x & 31;
    int wv   = threadIdx.x >> 5;
    if (lane == 0) wsum[wv] = s;
    __syncthreads();
    if (threadIdx.x == 0) {
        float bs = 0.0f;
#pragma unroll
        for (int k = 0; k < NTHREADS / 32; ++k) bs += wsum[k];
        partials[blockIdx.x] = bs;
    }
}

__global__ void __launch_bounds__(NTHREADS)
reduce_final(const float* __restrict__ partials, float* __restrict__ out) {
    float acc = 0.0f;
    for (int i = threadIdx.x; i < NBLOCKS; i += NTHREADS)
        acc += partials[i];
    float s = wave_sum_wmma(acc);
    __shared__ float wsum[NTHREADS / 32];
    int lane = threadIdx.x & 31;
    int wv   = threadIdx.x >> 5;
    if (lane == 0) wsum[wv] = s;
    __syncthreads();
    if (threadIdx.x == 0) {
        float total = 0.0f;
#pragma unroll
        for (int k = 0; k < NTHREADS / 32; ++k) total += wsum[k];
        out[0] = total * INV_N;
    }
}

extern "C" void kernel_launch(void* const* d_in, const int* in_sizes, int n_in,
                              void* d_out, int out_size, void* d_ws, size_t ws_size,
                              hipStream_t stream) {
    const float* input  = (const float*)d_in[0];
    const float* target = (const float*)d_in[1];
    const float* flow   = (const float*)d_in[2];
    float* partials = (float*)d_ws;   // NBLOCKS * 4 bytes = 16 KB scratch

    warp_l1_partial<<<NBLOCKS, NTHREADS, 0, stream>>>(input, target, flow, partials);
    reduce_final<<<1, NTHREADS, 0, stream>>>(partials, (float*)d_out);
}